// Adapter_19430432047214
// MI455X (gfx1250) — compile-verified
//
#include <hip/hip_runtime.h>
#include <math.h>

// Problem constants (match reference)
#define B_    32
#define C_    768
#define HW_   64
#define PIX   (HW_ * HW_)   // 4096
#define EMB_  256           // (64/4)^2
#define DOWN_ 64            // 768/12

typedef float v2f __attribute__((ext_vector_type(2)));
typedef float v8f __attribute__((ext_vector_type(8)));

// ---------------------------------------------------------------------------
// Kernel 1: 1x1 conv C->1  (channel reduction).  Reads all of x (402 MB).
// Each thread owns 4 consecutive pixels (float4); a wave reads 128 consecutive
// pixels per channel -> fully coalesced B128 loads; conv_w[c] is wave-uniform.
// ---------------------------------------------------------------------------
__global__ __launch_bounds__(256) void adapter_conv_reduce(
    const float* __restrict__ x, const float* __restrict__ conv_w,
    const float* __restrict__ conv_b, float* __restrict__ y) {
  int t     = blockIdx.x * blockDim.x + threadIdx.x;       // 32768 threads
  int b     = t / (PIX / 4);
  int pbase = (t % (PIX / 4)) * 4;
  const float* xb = x + (size_t)b * C_ * PIX + pbase;
  float4 acc = make_float4(0.f, 0.f, 0.f, 0.f);
  for (int c = 0; c < C_; ++c) {
    float wv  = conv_w[c];
    float4 xv = *(const float4*)(xb + (size_t)c * PIX);
    acc.x = fmaf(xv.x, wv, acc.x);
    acc.y = fmaf(xv.y, wv, acc.y);
    acc.z = fmaf(xv.z, wv, acc.z);
    acc.w = fmaf(xv.w, wv, acc.w);
  }
  float bb = conv_b[0];
  float4 o = make_float4(acc.x + bb, acc.y + bb, acc.z + bb, acc.w + bb);
  *(float4*)(y + (size_t)b * PIX + pbase) = o;
}

// ---------------------------------------------------------------------------
// Kernel 2: LayerNorm over W=64, then 4x4 maxpool with window-local argmax
// (first-max tie-break).  One block per batch, 64 threads (row per thread).
// ---------------------------------------------------------------------------
__global__ __launch_bounds__(64) void adapter_ln_pool(
    const float* __restrict__ y, const float* __restrict__ ln_g,
    const float* __restrict__ ln_b, float* __restrict__ flat,
    int* __restrict__ idx) {
  __shared__ float s[PIX];          // 16 KB
  int b = blockIdx.x;
  int t = threadIdx.x;              // 0..63
  const float* yb = y + (size_t)b * PIX;
  for (int i = t; i < PIX; i += 64) s[i] = yb[i];
  __syncthreads();
  // LayerNorm over row t
  float mu = 0.f;
  for (int w = 0; w < HW_; ++w) mu += s[t * HW_ + w];
  mu *= (1.f / HW_);
  float var = 0.f;
  for (int w = 0; w < HW_; ++w) {
    float d = s[t * HW_ + w] - mu;
    var = fmaf(d, d, var);
  }
  var *= (1.f / HW_);
  float rs = rsqrtf(var + 1e-5f);
  for (int w = 0; w < HW_; ++w)
    s[t * HW_ + w] = (s[t * HW_ + w] - mu) * rs * ln_g[w] + ln_b[w];
  __syncthreads();
  // 256 windows of 4x4; thread handles 4 windows
  for (int wi = t; wi < EMB_; wi += 64) {
    int hp = wi >> 4, wp = wi & 15;
    float best = -INFINITY;
    int   bi   = 0;
    for (int r = 0; r < 4; ++r)
      for (int c2 = 0; c2 < 4; ++c2) {
        float v = s[(hp * 4 + r) * HW_ + wp * 4 + c2];
        if (v > best) { best = v; bi = r * 4 + c2; }   // first-max tie-break
      }
    flat[b * EMB_ + wi] = best;
    idx [b * EMB_ + wi] = bi;
  }
}

// ---------------------------------------------------------------------------
// Kernel 3: bottleneck MLP via V_WMMA_F32_16X16X4_F32 (exact f32 semantics).
//   down = relu(flat[32,256] @ down_w[64,256]^T + down_b)   -> LDS
//   up   = down[32,64]       @ up_w[256,64]^T   + up_b      -> ws
// One 256-thread block = 8 waves.  Down: 8 tiles (1/wave).  Up: 32 tiles
// (4/wave).  Fragment layouts per CDNA5 ISA 7.12.2 (f32 A/B = 2 VGPR,
// C/D = 8 VGPR).
// ---------------------------------------------------------------------------
__global__ __launch_bounds__(256) void adapter_mlp_wmma(
    const float* __restrict__ flat,
    const float* __restrict__ down_w, const float* __restrict__ down_b,
    const float* __restrict__ up_w,   const float* __restrict__ up_b,
    float* __restrict__ up_out) {
  __shared__ float s_down[B_ * DOWN_];            // 8 KB
  const int lane  = threadIdx.x & 31;
  const int wv    = threadIdx.x >> 5;             // wave id 0..7
  const int lmod  = lane & 15;                    // M or N within tile
  const int khalf = (lane >> 4) << 1;             // 0 or 2
  const int mhalf = (lane >> 4) * 8;              // C/D row half

  // ---- down-projection: wave -> (mtile = wv/4, ntile = wv%4) ----
  {
    const int Mb = (wv >> 2) * 16, Nb = (wv & 3) * 16;
    v8f acc = {};
    const float* arow = flat   + (size_t)(Mb + lmod) * EMB_;
    const float* brow = down_w + (size_t)(Nb + lmod) * EMB_;  // B[k][n]=down_w[n][k]
    for (int k0 = 0; k0 < EMB_; k0 += 4) {
      v2f a = { arow[k0 + khalf], arow[k0 + khalf + 1] };
      v2f bq = { brow[k0 + khalf], brow[k0 + khalf + 1] };
      acc = __builtin_amdgcn_wmma_f32_16x16x4_f32(
          /*neg_a=*/false, a, /*neg_b=*/false, bq,
          /*c_mod=*/(short)0, acc, /*reuse_a=*/false, /*reuse_b=*/false);
    }
    const int   n    = Nb + lmod;
    const float bias = down_b[n];
#pragma unroll
    for (int v = 0; v < 8; ++v) {
      int m = Mb + mhalf + v;
      s_down[m * DOWN_ + n] = fmaxf(acc[v] + bias, 0.f);      // ReLU
    }
  }
  __syncthreads();

  // ---- up-projection: 32 tiles over 8 waves ----
  for (int j = 0; j < 4; ++j) {
    const int tt = wv * 4 + j;
    const int Mb = (tt >> 4) * 16, Nb = (tt & 15) * 16;
    v8f acc = {};
    const float* arow = s_down + (Mb + lmod) * DOWN_;
    const float* brow = up_w + (size_t)(Nb + lmod) * DOWN_;   // B[k][n]=up_w[n][k]
    for (int k0 = 0; k0 < DOWN_; k0 += 4) {
      v2f a = { arow[k0 + khalf], arow[k0 + khalf + 1] };
      v2f bq = { brow[k0 + khalf], brow[k0 + khalf + 1] };
      acc = __builtin_amdgcn_wmma_f32_16x16x4_f32(
          false, a, false, bq, (short)0, acc, false, false);
    }
    const int   n    = Nb + lmod;
    const float bias = up_b[n];
#pragma unroll
    for (int v = 0; v < 8; ++v) {
      int m = Mb + mhalf + v;
      up_out[m * EMB_ + n] = acc[v] + bias;
    }
  }
}

// ---------------------------------------------------------------------------
// Kernel 4: fused unpool + 1x1 deconv + residual.  The bandwidth kernel:
// read x (402 MB) + write out (402 MB), float4 per thread, fully coalesced.
// unp is computed on the fly from idx/up (2 KB per batch, cache-resident).
// ---------------------------------------------------------------------------
__global__ __launch_bounds__(256) void adapter_unpool_deconv_res(
    const float* __restrict__ x, const float* __restrict__ up,
    const int* __restrict__ idx, const float* __restrict__ deconv_w,
    const float* __restrict__ deconv_b, float* __restrict__ out) {
  size_t t = (size_t)blockIdx.x * blockDim.x + threadIdx.x;
  size_t e = t * 4;                         // element offset, 16B aligned
  int    pix = (int)(e & (PIX - 1));        // PIX = 4096
  size_t bc  = e >> 12;
  int    c   = (int)(bc % C_);
  int    b   = (int)(bc / C_);
  int h = pix >> 6, w = pix & 63;           // w is a multiple of 4
  int win = b * EMB_ + (h >> 2) * 16 + (w >> 2);
  int   iv = idx[win];
  float uv = up[win];
  float dw = deconv_w[c], db = deconv_b[c];
  int slot0 = (h & 3) * 4;                  // + (w&3), w&3 == 0
  float4 xv = *(const float4*)(x + e);
  float4 o;
  o.x = fmaf((iv == slot0 + 0) ? uv : 0.f, dw, xv.x + db);
  o.y = fmaf((iv == slot0 + 1) ? uv : 0.f, dw, xv.y + db);
  o.z = fmaf((iv == slot0 + 2) ? uv : 0.f, dw, xv.z + db);
  o.w = fmaf((iv == slot0 + 3) ? uv : 0.f, dw, xv.w + db);
  *(float4*)(out + e) = o;
}

// ---------------------------------------------------------------------------
extern "C" void kernel_launch(void* const* d_in, const int* in_sizes, int n_in,
                              void* d_out, int out_size, void* d_ws, size_t ws_size,
                              hipStream_t stream) {
  const float* x        = (const float*)d_in[0];
  const float* conv_w   = (const float*)d_in[1];
  const float* conv_b   = (const float*)d_in[2];
  const float* ln_g     = (const float*)d_in[3];
  const float* ln_b     = (const float*)d_in[4];
  const float* down_w   = (const float*)d_in[5];
  const float* down_b   = (const float*)d_in[6];
  const float* up_w     = (const float*)d_in[7];
  const float* up_b     = (const float*)d_in[8];
  const float* deconv_w = (const float*)d_in[9];
  const float* deconv_b = (const float*)d_in[10];
  float* out = (float*)d_out;

  // Workspace layout: y | flat | idx | up  (~608 KB)
  float* ws_y    = (float*)d_ws;                 // B*PIX    = 131072 f32
  float* ws_flat = ws_y + (size_t)B_ * PIX;      // B*EMB    = 8192 f32
  int*   ws_idx  = (int*)(ws_flat + (size_t)B_ * EMB_);
  float* ws_up   = (float*)(ws_idx + (size_t)B_ * EMB_);

  // 1) channel reduction: 32768 threads, each a float4 of pixels
  adapter_conv_reduce<<<(B_ * PIX / 4) / 256, 256, 0, stream>>>(
      x, conv_w, conv_b, ws_y);

  // 2) LayerNorm + maxpool(4,4) with indices: one block per batch
  adapter_ln_pool<<<B_, 64, 0, stream>>>(ws_y, ln_g, ln_b, ws_flat, ws_idx);

  // 3) bottleneck MLP on WMMA f32 path: single 8-wave block
  adapter_mlp_wmma<<<1, 256, 0, stream>>>(ws_flat, down_w, down_b,
                                          up_w, up_b, ws_up);

  // 4) unpool + deconv + residual: the 805 MB streaming kernel
  size_t n4 = (size_t)B_ * C_ * PIX / 4;         // 25,165,824 float4
  adapter_unpool_deconv_res<<<(unsigned)(n4 / 256), 256, 0, stream>>>(
      x, ws_up, ws_idx, deconv_w, deconv_b, out);
}